// CVQNN_61091614818842
// MI455X (gfx1250) — compile-verified
//
#include <hip/hip_runtime.h>
#include <hip/hip_bf16.h>

#define CUT 15
#define NLAYERS 5
#define BATCH 1048576
#define ROWS_PER_BLOCK 256
#define LDS_STRIDE 34   // 34 floats/row: keeps b64 LDS reads 8B-aligned, spreads banks

typedef __attribute__((ext_vector_type(2))) float v2f;
typedef __attribute__((ext_vector_type(8))) float v8f;

__device__ __forceinline__ float2 cmul(float2 a, float2 b) {
    return make_float2(a.x * b.x - a.y * b.y, a.x * b.y + a.y * b.x);
}
__device__ __forceinline__ float2 cadd(float2 a, float2 b) {
    return make_float2(a.x + b.x, a.y + b.y);
}
__device__ __forceinline__ float2 cscale(float s, float2 a) {
    return make_float2(s * a.x, s * a.y);
}

// ---------------------------------------------------------------------------
// Kernel 1: fold all 5 layer unitaries into one 15x15 complex matrix, emit it
// as a real 32x32 matrix W (complex -> 2x2 real blocks, zero padded) in d_ws.
//   out_flat[2m]   = sum_n  Re(U)[m,n]*psi_r[n] - Im(U)[m,n]*psi_i[n]
//   out_flat[2m+1] = sum_n  Im(U)[m,n]*psi_r[n] + Re(U)[m,n]*psi_i[n]
// ---------------------------------------------------------------------------
__global__ void build_w_kernel(const float* __restrict__ th1, const float* __restrict__ sr,
                               const float* __restrict__ sth, const float* __restrict__ th2,
                               const float* __restrict__ dr, const float* __restrict__ dphi,
                               const float* __restrict__ kap, float* __restrict__ W) {
    __shared__ float2 S[CUT][CUT], D[CUT][CUT], U[CUT][CUT], T[CUT][CUT], Tmp[CUT][CUT];
    const int tid = threadIdx.x;
    const int m = tid / CUT, n = tid % CUT;

    if (tid < CUT * CUT) T[m][n] = make_float2((m == n) ? 1.f : 0.f, 0.f);
    __syncthreads();

    for (int layer = 0; layer < NLAYERS; ++layer) {
        const float t1 = th1[layer], r = sr[layer], st = sth[layer];
        const float t2 = th2[layer], drr = dr[layer], dp = dphi[layer], kp = kap[layer];

        // ---- squeeze matrix S(z), z = r e^{i st} (thewalrus recurrence) ----
        const float tr = tanhf(r), sech = 1.f / coshf(r);
        const float2 Tq = make_float2(cosf(st) * tr, sinf(st) * tr); // e^{i st} tanh r
        const float2 Tc = make_float2(Tq.x, -Tq.y);
        if (tid == 0) {
            S[0][0] = make_float2(sqrtf(sech), 0.f);
            for (int mm = 1; mm < CUT; ++mm) S[mm][0] = make_float2(0.f, 0.f);
            for (int mm = 2; mm < CUT; mm += 2) {
                float f = sqrtf((float)(mm - 1)) / sqrtf((float)mm);
                S[mm][0] = cscale(-f, cmul(Tq, S[mm - 2][0]));
            }
        }
        __syncthreads();
        for (int nn = 1; nn < CUT; ++nn) {             // columns sequential, rows parallel
            if (tid < CUT) {
                const int mm = tid;
                float2 v = make_float2(0.f, 0.f);
                if (((mm + nn) & 1) == 0) {
                    const float isn = 1.f / sqrtf((float)nn);
                    if (nn >= 2) v = cscale(sqrtf((float)(nn - 1)) * isn, cmul(Tc, S[mm][nn - 2]));
                    if (mm >= 1) v = cadd(v, cscale(sqrtf((float)mm) * isn * sech, S[mm - 1][nn - 1]));
                }
                S[mm][nn] = v;
            }
            __syncthreads();
        }

        // ---- displacement matrix D(alpha), alpha = drr e^{i dp} ----
        const float2 alpha = make_float2(drr * cosf(dp), drr * sinf(dp));
        const float2 malc  = make_float2(-drr * cosf(dp), drr * sinf(dp)); // -conj(alpha)
        if (tid == 0) {
            D[0][0] = make_float2(expf(-0.5f * drr * drr), 0.f);
            for (int mm = 1; mm < CUT; ++mm)
                D[mm][0] = cscale(1.f / sqrtf((float)mm), cmul(alpha, D[mm - 1][0]));
        }
        __syncthreads();
        for (int nn = 1; nn < CUT; ++nn) {             // each column depends on previous only
            if (tid < CUT) {
                const int mm = tid;
                const float isn = 1.f / sqrtf((float)nn);
                float2 v = cscale(isn, cmul(malc, D[mm][nn - 1]));
                if (mm >= 1) v = cadd(v, cscale(sqrtf((float)mm) * isn, D[mm - 1][nn - 1]));
                D[mm][nn] = v;
            }
            __syncthreads();
        }

        // ---- U = diag(kv) . D . diag(p2) . S . diag(p1) ----
        if (tid < CUT * CUT) {
            float2 acc = make_float2(0.f, 0.f);
            for (int j = 0; j < CUT; ++j) {
                const float2 p2 = make_float2(cosf(t2 * (float)j), sinf(t2 * (float)j));
                acc = cadd(acc, cmul(D[m][j], cmul(p2, S[j][n])));
            }
            const float2 p1 = make_float2(cosf(t1 * (float)n), sinf(t1 * (float)n));
            const float  m2 = (float)(m * m);
            const float2 kv = make_float2(cosf(kp * m2), sinf(kp * m2));
            U[m][n] = cmul(kv, cmul(acc, p1));
        }
        __syncthreads();

        // ---- T = U @ T ----
        if (tid < CUT * CUT) {
            float2 acc = make_float2(0.f, 0.f);
            for (int j = 0; j < CUT; ++j) acc = cadd(acc, cmul(U[m][j], T[j][n]));
            Tmp[m][n] = acc;
        }
        __syncthreads();
        if (tid < CUT * CUT) T[m][n] = Tmp[m][n];
        __syncthreads();
    }

    // ---- expand to real 32x32, row-major: W[row][col], rows/cols >= 30 are zero ----
    for (int idx = tid; idx < 32 * 32; idx += blockDim.x) {
        const int rr = idx >> 5, cc = idx & 31;
        const int mm = rr >> 1, nn = cc >> 1;
        float v = 0.f;
        if (mm < CUT && nn < CUT) {
            const float2 u = T[mm][nn];
            const int a = rr & 1, b = cc & 1;
            v = (a == b) ? u.x : (a ? u.y : -u.y);  // [Re,-Im; Im,Re]
        }
        W[idx] = v;
    }
}

// ---------------------------------------------------------------------------
// Kernel 2: out(Bx30) = psi(Bx30) . W^T using V_WMMA_F32_16X16X4_F32.
// Per wave: two M=16 row tiles (32 rows), N=32 (two C tiles), K=32 in 8 steps.
// B fragments (constant W) loaded once per wave, reused for both row tiles.
// ---------------------------------------------------------------------------
__global__ __launch_bounds__(256) void apply_w_kernel(const float* __restrict__ psi,
                                                      const float* __restrict__ W,
                                                      float* __restrict__ out) {
    __shared__ float lds[ROWS_PER_BLOCK * LDS_STRIDE];
    const int tid  = threadIdx.x;
    const int lane = tid & 31;
    const int wave = tid >> 5;
    const long long blockRow = (long long)blockIdx.x * ROWS_PER_BLOCK;

    // Stage 256 rows x 30 floats -> LDS via b128 global loads (16B-aligned:
    // blockRow*120 bytes is a multiple of 16). 1920 float4 per block.
    {
        const float4* __restrict__ src = (const float4*)(psi + blockRow * 30);
        for (int idx = tid; idx < ROWS_PER_BLOCK * 30 / 4; idx += 256) {
            const float4 v = src[idx];
            const int f = idx << 2;
            lds[((f + 0) / 30) * LDS_STRIDE + (f + 0) % 30] = v.x;
            lds[((f + 1) / 30) * LDS_STRIDE + (f + 1) % 30] = v.y;
            lds[((f + 2) / 30) * LDS_STRIDE + (f + 2) % 30] = v.z;
            lds[((f + 3) / 30) * LDS_STRIDE + (f + 3) % 30] = v.w;
        }
    }
    // Zero K-padding columns 30..31 (32..33 never read)
    for (int idx = tid; idx < ROWS_PER_BLOCK * 2; idx += 256) {
        lds[(idx >> 1) * LDS_STRIDE + 30 + (idx & 1)] = 0.f;
    }

    // B fragments: B[k][n] = W[n][k]; ISA f32 4x16 layout:
    // lanes 0-15: N=lane, {K=k0, K=k0+1}; lanes 16-31: N=lane-16, {K=k0+2, K=k0+3}
    const int nlo   = lane & 15;
    const int khalf = (lane >> 4) << 1;     // 0 or 2
    v2f b0[8], b1[8];
#pragma unroll
    for (int s = 0; s < 8; ++s) {
        const int kk = 4 * s + khalf;
        b0[s].x = W[nlo * 32 + kk];
        b0[s].y = W[nlo * 32 + kk + 1];
        b1[s].x = W[(nlo + 16) * 32 + kk];
        b1[s].y = W[(nlo + 16) * 32 + kk + 1];
    }
    __syncthreads();

    // Two 16-row tiles per wave; A fragments from LDS (ISA f32 16x4 layout:
    // lanes 0-15: M=lane, {K=k0,k0+1}; lanes 16-31: M=lane-16, {K=k0+2,k0+3})
#pragma unroll
    for (int t = 0; t < 2; ++t) {
        const int rowBase = wave * 32 + t * 16;
        v8f c0 = {}; v8f c1 = {};
#pragma unroll
        for (int s = 0; s < 8; ++s) {
            const int kk = 4 * s + khalf;
            const float* p = &lds[(rowBase + nlo) * LDS_STRIDE + kk];
            v2f a; a.x = p[0]; a.y = p[1];
            c0 = __builtin_amdgcn_wmma_f32_16x16x4_f32(false, a, false, b0[s],
                                                       (short)0, c0, false, false);
            c1 = __builtin_amdgcn_wmma_f32_16x16x4_f32(false, a, false, b1[s],
                                                       (short)0, c1, false, false);
        }
        // C layout: VGPR i -> M = i (lanes 0-15) / i+8 (lanes 16-31), N = lane&15
        const long long r0 = blockRow + rowBase + (long long)((lane >> 4) << 3);
#pragma unroll
        for (int i = 0; i < 8; ++i) {
            out[(r0 + i) * 30 + nlo] = c0[i];            // cols 0..15
        }
        if (nlo < 14) {                                   // single branch per tile
#pragma unroll
            for (int i = 0; i < 8; ++i) {
                out[(r0 + i) * 30 + 16 + nlo] = c1[i];   // cols 16..29
            }
        }
    }
}

extern "C" void kernel_launch(void* const* d_in, const int* in_sizes, int n_in,
                              void* d_out, int out_size, void* d_ws, size_t ws_size,
                              hipStream_t stream) {
    const float* psi  = (const float*)d_in[0];   // (B,15) complex64 -> 30 floats/row
    const float* th1  = (const float*)d_in[1];
    const float* sr   = (const float*)d_in[2];
    const float* sth  = (const float*)d_in[3];
    const float* th2  = (const float*)d_in[4];
    const float* dr   = (const float*)d_in[5];
    const float* dphi = (const float*)d_in[6];
    const float* kap  = (const float*)d_in[7];
    float* W = (float*)d_ws;                     // 32x32 floats = 4 KB scratch
    float* out = (float*)d_out;

    build_w_kernel<<<1, 256, 0, stream>>>(th1, sr, sth, th2, dr, dphi, kap, W);
    apply_w_kernel<<<BATCH / ROWS_PER_BLOCK, 256, 0, stream>>>(psi, W, out);
}